// AFM_81140522156064
// MI455X (gfx1250) — compile-verified
//
#include <hip/hip_runtime.h>
#include <hip/hip_bf16.h>
#include <math.h>

// AFM forward for MI455X (gfx1250, wave32).
// One workgroup (256 threads = 8 waves) per batch row.
// Attention-score GEMM uses V_WMMA_F32_16X16X4_F32 (fp32 in/out, matches
// the fp32 reference; problem is ~8.8 GFLOP total -> latency/compute bound,
// no reason to downcast).

typedef __attribute__((ext_vector_type(2))) float v2f;
typedef __attribute__((ext_vector_type(8))) float v8f;

#define FDIM 64
#define KDIM 64
#define PSTR 65   // padded LDS row stride (floats) to avoid bank conflicts

__global__ __launch_bounds__(256) void afm_fwd_kernel(
    const float* __restrict__ x,      // [B,64]
    const float* __restrict__ Ww,     // [64]
    const float* __restrict__ bw,     // [1]
    const float* __restrict__ cross,  // [64,64]
    const float* __restrict__ W1,     // [64,64]
    const float* __restrict__ b1,     // [64]
    const float* __restrict__ w2,     // [64]
    const float* __restrict__ Wd1,    // [32,64]
    const float* __restrict__ bd1,    // [32]
    const float* __restrict__ Wd2,    // [21,32]
    const float* __restrict__ bd2,    // [21]
    const float* __restrict__ Wd3,    // [16,21]
    const float* __restrict__ bd3,    // [16]
    const float* __restrict__ Wo,     // [16]
    const float* __restrict__ bo,     // [1]
    float* __restrict__ out)          // [B]
{
    __shared__ float sP[FDIM * PSTR];   // p[i][k] = x[i]*cross[i][k]
    __shared__ float sW1[KDIM * PSTR];  // W1[l][k]
    __shared__ float sE[FDIM * FDIM];   // scores, then exp-weights
    __shared__ float sX[FDIM];
    __shared__ float sB1[KDIM];
    __shared__ float sW2[KDIM];
    __shared__ float sTmp[4 * KDIM];
    __shared__ float sT[KDIM];
    __shared__ float sH1[32];
    __shared__ float sH2[21];
    __shared__ float sH3[16];
    __shared__ float sRed[8];
    __shared__ float sScal[2];          // [0]=softmax max m, [1]=denom

    const int b    = blockIdx.x;
    const int t    = threadIdx.x;
    const int lane = t & 31;
    const int wave = t >> 5;

    // ---- stage 0: x row + small vectors into LDS ----
    if (t < FDIM) {
        sX[t]  = x[b * FDIM + t];
        sB1[t] = b1[t];
        sW2[t] = w2[t];
    }
    __syncthreads();

    // ---- stage 1: p = x_i * cross_i and W1 into LDS (padded rows) ----
    for (int idx = t; idx < FDIM * KDIM; idx += 256) {
        int i = idx >> 6, k = idx & 63;
        sP[i * PSTR + k]  = sX[i] * cross[idx];
        sW1[i * PSTR + k] = W1[idx];
    }
    __syncthreads();

    // ---- stage 2: attention scores s[i*64+j] via WMMA fp32 16x16x4 ----
    // GEMM: H[pair, l] = sum_k V[pair,k] * W1[l,k],  V[pair,k]=p_i[k]*p_j[k]
    {
        const int m  = lane & 15;   // A-row (pair within tile) / B-column
        const int hi = lane >> 4;   // selects K sub-pair {0,1} vs {2,3}

        for (int ptl = 0; ptl < 32; ++ptl) {           // 8 waves x 32 = 256 tiles
            const int pbase = (wave * 32 + ptl) * 16;  // 16 pairs per tile
            const int pr = pbase + m;
            const int pi = pr >> 6;                    // constant across tile
            const int pj = pr & 63;                    // consecutive across lanes
            const float* rowi = &sP[pi * PSTR];
            const float* rowj = &sP[pj * PSTR];

            v8f acc[4] = {};   // 4 l-tiles of 16 channels
            #pragma unroll
            for (int kc = 0; kc < 16; ++kc) {
                const int k0 = kc * 4 + hi * 2;
                v2f a;
                a.x = rowi[k0]     * rowj[k0];
                a.y = rowi[k0 + 1] * rowj[k0 + 1];
                #pragma unroll
                for (int lt = 0; lt < 4; ++lt) {
                    const int n = lt * 16 + m;
                    v2f bb;
                    bb.x = sW1[n * PSTR + k0];
                    bb.y = sW1[n * PSTR + k0 + 1];
                    acc[lt] = __builtin_amdgcn_wmma_f32_16x16x4_f32(
                        false, a, false, bb, (short)0, acc[lt], false, false);
                }
            }

            // relu(H + b1) . w2  reduced over the 64 l-channels
            float part[8];
            #pragma unroll
            for (int v = 0; v < 8; ++v) part[v] = 0.f;
            #pragma unroll
            for (int lt = 0; lt < 4; ++lt) {
                const int ng = lt * 16 + m;       // this lane's l-channel
                const float bb1 = sB1[ng], ww2 = sW2[ng];
                #pragma unroll
                for (int v = 0; v < 8; ++v) {
                    float h = acc[lt][v] + bb1;
                    h = h > 0.f ? h : 0.f;
                    part[v] += h * ww2;
                }
            }
            // reduce across the 16 lanes of each half-wave (n dimension)
            #pragma unroll
            for (int v = 0; v < 8; ++v) {
                #pragma unroll
                for (int off = 1; off < 16; off <<= 1)
                    part[v] += __shfl_xor(part[v], off, 32);
            }
            if (m == 0) {
                const int mb = pbase + hi * 8;    // half 0 -> rows 0..7, half 1 -> 8..15
                #pragma unroll
                for (int v = 0; v < 8; ++v) sE[mb + v] = part[v];
            }
        }
    }
    __syncthreads();

    // ---- stage 3: softmax max over valid (upper-tri, nonzero x) pairs ----
    {
        float lmax = -INFINITY;
        const int base = t * 16;
        for (int q = 0; q < 16; ++q) {
            const int pr = base + q;
            const int i = pr >> 6, j = pr & 63;
            const bool valid = (j >= i) && (sX[i] != 0.f) && (sX[j] != 0.f);
            if (valid) lmax = fmaxf(lmax, sE[pr]);
        }
        #pragma unroll
        for (int off = 1; off < 32; off <<= 1)
            lmax = fmaxf(lmax, __shfl_xor(lmax, off, 32));
        if (lane == 0) sRed[wave] = lmax;
    }
    __syncthreads();
    if (t == 0) {
        float mm = 0.f;  // 0-logit seed entry participates in the max
        for (int w = 0; w < 8; ++w) mm = fmaxf(mm, sRed[w]);
        sScal[0] = mm;
    }
    __syncthreads();

    // ---- stage 4: e = exp(s - m) for valid pairs, and denom ----
    {
        const float mm = sScal[0];
        float lsum = 0.f;
        const int base = t * 16;
        for (int q = 0; q < 16; ++q) {
            const int pr = base + q;
            const int i = pr >> 6, j = pr & 63;
            const bool valid = (j >= i) && (sX[i] != 0.f) && (sX[j] != 0.f);
            const float e = valid ? expf(sE[pr] - mm) : 0.f;
            sE[pr] = e;
            lsum += e;
        }
        #pragma unroll
        for (int off = 1; off < 32; off <<= 1)
            lsum += __shfl_xor(lsum, off, 32);
        if (lane == 0) sRed[wave] = lsum;
    }
    __syncthreads();
    if (t == 0) {
        float d = expf(-sScal[0]);  // the zero-logit seed term
        for (int w = 0; w < 8; ++w) d += sRed[w];
        sScal[1] = d;
    }
    __syncthreads();

    // ---- stage 5: tmp[k] = sum_ij e[i,j] * p_i[k]*p_j[k] / denom ----
    {
        const int k  = t & 63;
        const int ic = t >> 6;            // 4 chunks of 16 i-rows
        float accv = 0.f;
        for (int ii = 0; ii < 16; ++ii) {
            const int i = ic * 16 + ii;
            float inner = 0.f;
            for (int j = 0; j < 64; ++j)
                inner += sE[i * 64 + j] * sP[j * PSTR + k];
            accv += sP[i * PSTR + k] * inner;
        }
        sTmp[ic * 64 + k] = accv;
    }
    __syncthreads();
    if (t < 64)
        sT[t] = (sTmp[t] + sTmp[64 + t] + sTmp[128 + t] + sTmp[192 + t]) / sScal[1];
    __syncthreads();

    // ---- stage 6: deep MLP 64->32->21->16->1, + linear, sigmoid ----
    if (t < 32) {
        float a = bd1[t];
        for (int k2 = 0; k2 < 64; ++k2) a += Wd1[t * 64 + k2] * sT[k2];
        sH1[t] = a > 0.f ? a : 0.f;
    }
    __syncthreads();
    if (t < 21) {
        float a = bd2[t];
        for (int k2 = 0; k2 < 32; ++k2) a += Wd2[t * 32 + k2] * sH1[k2];
        sH2[t] = a > 0.f ? a : 0.f;
    }
    __syncthreads();
    if (t < 16) {
        float a = bd3[t];
        for (int k2 = 0; k2 < 21; ++k2) a += Wd3[t * 21 + k2] * sH2[k2];
        sH3[t] = a > 0.f ? a : 0.f;
    }
    __syncthreads();
    if (t == 0) {
        float d = bo[0];
        for (int k2 = 0; k2 < 16; ++k2) d += Wo[k2] * sH3[k2];
        float lin = bw[0];
        for (int f = 0; f < 64; ++f) lin += sX[f] * Ww[f];
        const float z = lin + d;
        out[b] = 1.f / (1.f + expf(-z));
    }
}

extern "C" void kernel_launch(void* const* d_in, const int* in_sizes, int n_in,
                              void* d_out, int out_size, void* d_ws, size_t ws_size,
                              hipStream_t stream) {
    const float* x     = (const float*)d_in[0];
    const float* Ww    = (const float*)d_in[1];
    const float* bw    = (const float*)d_in[2];
    const float* cross = (const float*)d_in[3];
    const float* W1    = (const float*)d_in[4];
    const float* b1    = (const float*)d_in[5];
    const float* w2    = (const float*)d_in[6];
    const float* Wd1   = (const float*)d_in[7];
    const float* bd1   = (const float*)d_in[8];
    const float* Wd2   = (const float*)d_in[9];
    const float* bd2   = (const float*)d_in[10];
    const float* Wd3   = (const float*)d_in[11];
    const float* bd3   = (const float*)d_in[12];
    const float* Wo    = (const float*)d_in[13];
    const float* bo    = (const float*)d_in[14];
    float* out = (float*)d_out;

    const int B = out_size;  // one output per batch row
    afm_fwd_kernel<<<B, 256, 0, stream>>>(x, Ww, bw, cross, W1, b1, w2,
                                          Wd1, bd1, Wd2, bd2, Wd3, bd3,
                                          Wo, bo, out);
    (void)d_ws; (void)ws_size; (void)n_in; (void)in_sizes;
}